// HistogramLoss_40140764348889
// MI455X (gfx1250) — compile-verified
//
#include <hip/hip_runtime.h>

// ---- problem constants (fixed by the reference) ----
#define BINS 256
#define NBATCH 32
#define NCHAN 3
#define SLICES (NBATCH * NCHAN)     // 96 (b,c) slices
#define HW (512 * 512)              // 262144 elements per slice
#define THREADS 256                 // 8 wave32 waves per block
#define NSUB 8                      // one private LDS histogram per wave
#define CHUNKS_PER_SLICE 32
#define CHUNK_ELEMS (HW / CHUNKS_PER_SLICE)          // 8192
#define VEC_ITERS (CHUNK_ELEMS / 4 / THREADS)        // 8 vec4 loads / thread

// raw clang ext-vector: accepted by __builtin_nontemporal_load (HIP float4 is not)
typedef __attribute__((ext_vector_type(4))) float v4f;

__global__ void HistLoss_zero_u32(unsigned* __restrict__ p, int n) {
    int i = blockIdx.x * blockDim.x + threadIdx.x;
    if (i < n) p[i] = 0u;
}

__device__ __forceinline__ int binof(float x) {
    // reference: clip(int32(x*256), 0, 255); trunc == floor for x >= 0
    int b = (int)(x * 256.0f);
    b = b < 0 ? 0 : b;
    return b > (BINS - 1) ? (BINS - 1) : b;
}

// One block handles a 8192-element chunk of one (b,c) slice.
// Per-wave private LDS histograms -> ds_add_u32 with zero inter-wave contention,
// merged once and flushed with one global atomic per bin per block.
__global__ __launch_bounds__(THREADS) void HistLoss_hist_kernel(
        const float* __restrict__ img, unsigned* __restrict__ counts) {
    __shared__ unsigned lhist[NSUB][BINS];   // 8 KB (vs 320 KB/WGP)
    const int t   = threadIdx.x;
    const int sub = t >> 5;                  // wave id within block (wave32)

    #pragma unroll
    for (int i = t; i < NSUB * BINS; i += THREADS)
        ((unsigned*)lhist)[i] = 0u;
    __syncthreads();

    const int slice = blockIdx.x / CHUNKS_PER_SLICE;
    const int chunk = blockIdx.x % CHUNKS_PER_SLICE;
    const v4f* __restrict__ p =
        reinterpret_cast<const v4f*>(img) +
        (size_t)slice * (HW / 4) + (size_t)chunk * (CHUNK_ELEMS / 4);

    #pragma unroll
    for (int it = 0; it < VEC_ITERS; ++it) {
        if (it + 2 < VEC_ITERS)
            __builtin_prefetch(&p[(it + 2) * THREADS + t], 0, 0);    // global_prefetch_b8
        // streamed once -> non-temporal b128 load, keep L2 for the other tensor
        v4f v = __builtin_nontemporal_load(&p[it * THREADS + t]);    // global_load_b128 th:NT
        atomicAdd(&lhist[sub][binof(v.x)], 1u);   // ds_add_u32
        atomicAdd(&lhist[sub][binof(v.y)], 1u);
        atomicAdd(&lhist[sub][binof(v.z)], 1u);
        atomicAdd(&lhist[sub][binof(v.w)], 1u);
    }
    __syncthreads();

    unsigned s = 0;
    #pragma unroll
    for (int i = 0; i < NSUB; ++i) s += lhist[i][t];
    atomicAdd(&counts[slice * BINS + t], s);       // global_atomic_add_u32
}

// Fold 32-batch signed count differences per (c,bin), apply |.| and exact
// normalization: each hist sums to exactly HW, batch mean /32, L1 mean /768.
__global__ __launch_bounds__(THREADS) void HistLoss_loss_kernel(
        const unsigned* __restrict__ cf, const unsigned* __restrict__ cr,
        float* __restrict__ out) {
    __shared__ float red[THREADS];
    const int t = threadIdx.x;
    const float inv = 1.0f / (32.0f * 262144.0f);
    float acc = 0.0f;
    #pragma unroll
    for (int k = 0; k < NCHAN; ++k) {
        const int pidx = k * THREADS + t;          // (c,bin) in [0,768)
        int d = 0;
        #pragma unroll 4
        for (int b = 0; b < NBATCH; ++b) {
            const int idx = b * (NCHAN * BINS) + pidx;
            d += (int)cf[idx] - (int)cr[idx];
        }
        acc += fabsf((float)d) * inv;
    }
    red[t] = acc;
    __syncthreads();
    #pragma unroll
    for (int s = THREADS / 2; s > 0; s >>= 1) {
        if (t < s) red[t] += red[t + s];
        __syncthreads();
    }
    if (t == 0) out[0] = red[0] * (1.0f / 768.0f);
}

extern "C" void kernel_launch(void* const* d_in, const int* in_sizes, int n_in,
                              void* d_out, int out_size, void* d_ws, size_t ws_size,
                              hipStream_t stream) {
    const float* fake = (const float*)d_in[0];
    const float* real = (const float*)d_in[1];
    unsigned* cf = (unsigned*)d_ws;                 // [32][3][256] u32 counts (fake)
    unsigned* cr = cf + SLICES * BINS;              // [32][3][256] u32 counts (real)
    float* out = (float*)d_out;

    const int ntot = 2 * SLICES * BINS;             // 49152 u32 = 192 KB of d_ws
    HistLoss_zero_u32<<<(ntot + THREADS - 1) / THREADS, THREADS, 0, stream>>>(cf, ntot);

    const dim3 grid(SLICES * CHUNKS_PER_SLICE);     // 3072 blocks
    HistLoss_hist_kernel<<<grid, THREADS, 0, stream>>>(fake, cf);
    HistLoss_hist_kernel<<<grid, THREADS, 0, stream>>>(real, cr);

    HistLoss_loss_kernel<<<1, THREADS, 0, stream>>>(cf, cr, out);
}